// PointTransformerBlock_32169305047427
// MI455X (gfx1250) — compile-verified
//
#include <hip/hip_runtime.h>

// ---------------------------------------------------------------------------
// PointTransformer block, MI455X (gfx1250, wave32).
// Bandwidth-bound pipeline: ~0.5 GB intermediate traffic => ~25us @ 23.3TB/s.
// GEMMs (1.3 GFLOP total) use exact-precision V_WMMA_F32_16X16X4_F32.
// Never materializes [E,64] tensors: 'a' and 'delta' recomputed from t3[E,3]
// plus L2-resident k/q/v (8 MB each << 192 MB L2).
// ---------------------------------------------------------------------------

#define NPTS 32768
#define KNB  16
#define EDG  (NPTS * KNB)

typedef float v2f __attribute__((ext_vector_type(2)));
typedef float v8f __attribute__((ext_vector_type(8)));

// ===========================================================================
// WMMA GEMM:  Y[n,o] = sum_i X[n,i] * W[o,i]  (+ bias[o])
// X:[rows,64] W:[64,64] row-major. Block = 64 rows, 4 waves (16 rows/wave).
// Each wave: 4 column tiles x 16 K-steps of v_wmma_f32_16x16x4_f32.
// Optionally accumulates per-column sum / sum-of-squares for BatchNorm stats.
// ===========================================================================
__global__ __launch_bounds__(128)
void gemm64_nt_kernel(const float* __restrict__ X, const float* __restrict__ W,
                      const float* __restrict__ bias, float* __restrict__ Y,
                      float* __restrict__ ssum, float* __restrict__ ssq) {
  __shared__ __attribute__((aligned(16))) float sW[64 * 64];
  __shared__ __attribute__((aligned(16))) float sX[64 * 64];
  __shared__ float sSum[64], sSq[64];

  const int t = threadIdx.x;
  const int blockRow = blockIdx.x * 64;

  for (int i = t * 4; i < 64 * 64; i += 128 * 4) {
    *(float4*)&sW[i] = *(const float4*)&W[i];
    *(float4*)&sX[i] = *(const float4*)&X[(size_t)blockRow * 64 + i];
  }
  if (ssum && t < 64) { sSum[t] = 0.f; sSq[t] = 0.f; }
  __syncthreads();

  const int lane  = t & 31;
  const int m     = lane & 15;     // row-in-tile (A) / col-in-tile (B,D)
  const int hi    = lane >> 4;     // half-wave selects K pair / M+8 rows
  const int rbase = (t >> 5) * 16; // wave's 16-row strip

  v8f acc[4] = {v8f{}, v8f{}, v8f{}, v8f{}};

  #pragma unroll
  for (int kk = 0; kk < 64; kk += 4) {
    const int col = kk + 2 * hi;
    // A 16x4 f32 layout: lanes0-15 {K=kk,kk+1}, lanes16-31 {K=kk+2,kk+3}
    v2f a = *(const v2f*)&sX[(rbase + m) * 64 + col];
    #pragma unroll
    for (int j = 0; j < 4; ++j) {
      // B 4x16: B[k][o] = W[o][k]; mirrored K striping across lane halves
      v2f b = *(const v2f*)&sW[(j * 16 + m) * 64 + col];
      acc[j] = __builtin_amdgcn_wmma_f32_16x16x4_f32(
          false, a, false, b, (short)0, acc[j], false, false);
    }
  }

  // D layout: vgpr r -> row (r + 8*hi), lane m -> column
  const int gRow0 = blockRow + rbase + 8 * hi;
  #pragma unroll
  for (int j = 0; j < 4; ++j) {
    const int c = j * 16 + m;
    const float bv = bias ? bias[c] : 0.f;
    float s = 0.f, q = 0.f;
    #pragma unroll
    for (int r = 0; r < 8; ++r) {
      float val = acc[j][r] + bv;
      Y[(size_t)(gRow0 + r) * 64 + c] = val;
      s += val; q += val * val;
    }
    if (ssum) { atomicAdd(&sSum[c], s); atomicAdd(&sSq[c], q); }
  }
  if (ssum) {
    __syncthreads();
    if (t < 64) { atomicAdd(&ssum[t], sSum[t]); atomicAdd(&ssq[t], sSq[t]); }
  }
}

// ===========================================================================
// BN finalize: scale = g*rsqrt(var+eps), off = b - mean*scale  (biased var)
// ===========================================================================
__global__ void finalize_kernel(const float* __restrict__ ssum,
                                const float* __restrict__ ssq,
                                const float* __restrict__ g,
                                const float* __restrict__ b,
                                float invN, int cols,
                                float* __restrict__ sc, float* __restrict__ of) {
  int c = threadIdx.x;
  if (c < cols) {
    float m = ssum[c] * invN;
    float v = ssq[c] * invN - m * m;
    float s = g[c] * rsqrtf(v + 1e-5f);
    sc[c] = s;
    of[c] = b[c] - m * s;
  }
}

// elementwise y = relu(x*scale[c] + off[c]), layout [rows,64]
__global__ __launch_bounds__(256)
void bnrelu_kernel(const float* __restrict__ Yin, const float* __restrict__ sc,
                   const float* __restrict__ of, float* __restrict__ H) {
  int i = blockIdx.x * 256 + threadIdx.x;
  int c = i & 63;
  float v = Yin[i] * sc[c] + of[c];
  H[i] = v > 0.f ? v : 0.f;
}

// ===========================================================================
// Edge pass 1: t3 = (pos[src]-pos[dst]) @ p1^T + p1_b ; column stats over E
// ===========================================================================
__global__ __launch_bounds__(256)
void edge_t3_kernel(const float* __restrict__ pos, const int* __restrict__ ei,
                    const float* __restrict__ p1w, const float* __restrict__ p1b,
                    float* __restrict__ t3, float* __restrict__ ssum,
                    float* __restrict__ ssq) {
  int e = blockIdx.x * 256 + threadIdx.x;
  int src = ei[e];
  int dst = e >> 4;  // edges are contiguous per target by construction
  float rx = pos[src * 3 + 0] - pos[dst * 3 + 0];
  float ry = pos[src * 3 + 1] - pos[dst * 3 + 1];
  float rz = pos[src * 3 + 2] - pos[dst * 3 + 2];
  float t0 = p1w[0] * rx + p1w[1] * ry + p1w[2] * rz + p1b[0];
  float t1 = p1w[3] * rx + p1w[4] * ry + p1w[5] * rz + p1b[1];
  float t2 = p1w[6] * rx + p1w[7] * ry + p1w[8] * rz + p1b[2];
  t3[(size_t)e * 3 + 0] = t0;
  t3[(size_t)e * 3 + 1] = t1;
  t3[(size_t)e * 3 + 2] = t2;

  float v0 = t0, v1 = t1, v2 = t2;
  float q0 = t0 * t0, q1 = t1 * t1, q2 = t2 * t2;
  #pragma unroll
  for (int d = 1; d < 32; d <<= 1) {
    v0 += __shfl_xor(v0, d); v1 += __shfl_xor(v1, d); v2 += __shfl_xor(v2, d);
    q0 += __shfl_xor(q0, d); q1 += __shfl_xor(q1, d); q2 += __shfl_xor(q2, d);
  }
  __shared__ float shS[8], shQ[8];
  if (threadIdx.x < 8) { shS[threadIdx.x] = 0.f; shQ[threadIdx.x] = 0.f; }
  __syncthreads();
  if ((threadIdx.x & 31) == 0) {
    atomicAdd(&shS[0], v0); atomicAdd(&shS[1], v1); atomicAdd(&shS[2], v2);
    atomicAdd(&shQ[0], q0); atomicAdd(&shQ[1], q1); atomicAdd(&shQ[2], q2);
  }
  __syncthreads();
  if (threadIdx.x < 3) {
    atomicAdd(&ssum[threadIdx.x], shS[threadIdx.x]);
    atomicAdd(&ssq[threadIdx.x], shQ[threadIdx.x]);
  }
}

// helper macro body shared by the two edge passes that need a = k[src]-q[dst]+delta
// wave-per-edge, 2 channels per lane (c0=2*lane, c1=2*lane+1) -> coalesced rows

// ===========================================================================
// Edge pass 2: column stats of a = k[src]-q[dst]+delta over E (64 cols)
// ===========================================================================
__global__ __launch_bounds__(256)
void edge_astats_kernel(const float* __restrict__ kb, const float* __restrict__ qb,
                        const float* __restrict__ t3,
                        const float* __restrict__ sc3, const float* __restrict__ of3,
                        const float* __restrict__ p2w, const float* __restrict__ p2b,
                        const int* __restrict__ ei,
                        float* __restrict__ ssum, float* __restrict__ ssq) {
  const int lane = threadIdx.x & 31;
  const int waveId = (blockIdx.x * blockDim.x + threadIdx.x) >> 5;
  const int nWaves = (gridDim.x * blockDim.x) >> 5;
  const int c0 = lane * 2, c1 = c0 + 1;
  const float pA0 = p2w[c0 * 3], pA1 = p2w[c0 * 3 + 1], pA2 = p2w[c0 * 3 + 2], pB0 = p2b[c0];
  const float pC0 = p2w[c1 * 3], pC1 = p2w[c1 * 3 + 1], pC2 = p2w[c1 * 3 + 2], pB1 = p2b[c1];
  const float s30 = sc3[0], s31 = sc3[1], s32 = sc3[2];
  const float o30 = of3[0], o31 = of3[1], o32 = of3[2];
  float s0 = 0, s1 = 0, q0 = 0, q1 = 0;
  for (int e = waveId; e < EDG; e += nWaves) {
    int src = ei[e];
    int dst = e >> 4;
    float d0 = fmaxf(t3[(size_t)e * 3 + 0] * s30 + o30, 0.f);
    float d1 = fmaxf(t3[(size_t)e * 3 + 1] * s31 + o31, 0.f);
    float d2 = fmaxf(t3[(size_t)e * 3 + 2] * s32 + o32, 0.f);
    float a0 = kb[(size_t)src * 64 + c0] - qb[(size_t)dst * 64 + c0]
             + (pA0 * d0 + pA1 * d1 + pA2 * d2 + pB0);
    float a1 = kb[(size_t)src * 64 + c1] - qb[(size_t)dst * 64 + c1]
             + (pC0 * d0 + pC1 * d1 + pC2 * d2 + pB1);
    s0 += a0; q0 += a0 * a0; s1 += a1; q1 += a1 * a1;
  }
  __shared__ float shs[64], shq[64];
  if (threadIdx.x < 64) { shs[threadIdx.x] = 0.f; shq[threadIdx.x] = 0.f; }
  __syncthreads();
  atomicAdd(&shs[c0], s0); atomicAdd(&shs[c1], s1);
  atomicAdd(&shq[c0], q0); atomicAdd(&shq[c1], q1);
  __syncthreads();
  if (threadIdx.x < 64) {
    atomicAdd(&ssum[threadIdx.x], shs[threadIdx.x]);
    atomicAdd(&ssq[threadIdx.x], shq[threadIdx.x]);
  }
}

// ===========================================================================
// Edge pass 3: a2[e,s] = relu(bn1(a)) @ w1^T + w1_b  (store [E,8], stats8)
// cross-lane 64->8 dot via shfl_xor butterfly
// ===========================================================================
__global__ __launch_bounds__(256)
void edge_a2_kernel(const float* __restrict__ kb, const float* __restrict__ qb,
                    const float* __restrict__ t3,
                    const float* __restrict__ sc3, const float* __restrict__ of3,
                    const float* __restrict__ sc64, const float* __restrict__ of64,
                    const float* __restrict__ p2w, const float* __restrict__ p2b,
                    const float* __restrict__ w1w, const float* __restrict__ w1b,
                    const int* __restrict__ ei, float* __restrict__ a2,
                    float* __restrict__ ssum, float* __restrict__ ssq) {
  const int lane = threadIdx.x & 31;
  const int waveId = (blockIdx.x * blockDim.x + threadIdx.x) >> 5;
  const int nWaves = (gridDim.x * blockDim.x) >> 5;
  const int c0 = lane * 2, c1 = c0 + 1;
  const float pA0 = p2w[c0 * 3], pA1 = p2w[c0 * 3 + 1], pA2 = p2w[c0 * 3 + 2], pB0 = p2b[c0];
  const float pC0 = p2w[c1 * 3], pC1 = p2w[c1 * 3 + 1], pC2 = p2w[c1 * 3 + 2], pB1 = p2b[c1];
  const float s30 = sc3[0], s31 = sc3[1], s32 = sc3[2];
  const float o30 = of3[0], o31 = of3[1], o32 = of3[2];
  const float scA = sc64[c0], ofA = of64[c0];
  const float scB = sc64[c1], ofB = of64[c1];
  float w1a[8], w1c[8];
  #pragma unroll
  for (int s = 0; s < 8; ++s) { w1a[s] = w1w[s * 64 + c0]; w1c[s] = w1w[s * 64 + c1]; }
  float bsum[8], bsq[8];
  #pragma unroll
  for (int s = 0; s < 8; ++s) { bsum[s] = 0.f; bsq[s] = 0.f; }

  for (int e = waveId; e < EDG; e += nWaves) {
    int src = ei[e];
    int dst = e >> 4;
    float d0 = fmaxf(t3[(size_t)e * 3 + 0] * s30 + o30, 0.f);
    float d1 = fmaxf(t3[(size_t)e * 3 + 1] * s31 + o31, 0.f);
    float d2 = fmaxf(t3[(size_t)e * 3 + 2] * s32 + o32, 0.f);
    float a0 = kb[(size_t)src * 64 + c0] - qb[(size_t)dst * 64 + c0]
             + (pA0 * d0 + pA1 * d1 + pA2 * d2 + pB0);
    float a1 = kb[(size_t)src * 64 + c1] - qb[(size_t)dst * 64 + c1]
             + (pC0 * d0 + pC1 * d1 + pC2 * d2 + pB1);
    float aw0 = fmaxf(a0 * scA + ofA, 0.f);
    float aw1 = fmaxf(a1 * scB + ofB, 0.f);
    float p[8];
    #pragma unroll
    for (int s = 0; s < 8; ++s) p[s] = aw0 * w1a[s] + aw1 * w1c[s];
    #pragma unroll
    for (int d = 1; d < 32; d <<= 1) {
      #pragma unroll
      for (int s = 0; s < 8; ++s) p[s] += __shfl_xor(p[s], d);
    }
    if (lane == 0) {
      #pragma unroll
      for (int s = 0; s < 8; ++s) {
        float val = p[s] + w1b[s];
        a2[(size_t)e * 8 + s] = val;
        bsum[s] += val; bsq[s] += val * val;
      }
    }
  }
  __shared__ float shs[8], shq[8];
  if (threadIdx.x < 8) { shs[threadIdx.x] = 0.f; shq[threadIdx.x] = 0.f; }
  __syncthreads();
  if (lane == 0) {
    #pragma unroll
    for (int s = 0; s < 8; ++s) { atomicAdd(&shs[s], bsum[s]); atomicAdd(&shq[s], bsq[s]); }
  }
  __syncthreads();
  if (threadIdx.x < 8) {
    atomicAdd(&ssum[threadIdx.x], shs[threadIdx.x]);
    atomicAdd(&ssq[threadIdx.x], shq[threadIdx.x]);
  }
}

// ===========================================================================
// Node pass: a3 = relu(bn2(a2)) @ w2^T + w2_b ; segment softmax over the 16
// contiguous edges (lane butterflies); out = sum_e attn[e,c&7]*(v[src]+delta).
// Wave-per-node. Fused bn2 statistics.
// ===========================================================================
__global__ __launch_bounds__(256)
void node_aggr_kernel(const float* __restrict__ a2,
                      const float* __restrict__ sc8, const float* __restrict__ of8,
                      const float* __restrict__ w2w, const float* __restrict__ w2b,
                      const float* __restrict__ t3,
                      const float* __restrict__ sc3, const float* __restrict__ of3,
                      const float* __restrict__ p2w, const float* __restrict__ p2b,
                      const float* __restrict__ vb, const int* __restrict__ ei,
                      float* __restrict__ outb,
                      float* __restrict__ ssum, float* __restrict__ ssq) {
  const int t = threadIdx.x;
  const int lane = t & 31;
  const int wv = t >> 5;
  __shared__ float sAttn[8][16][8];
  __shared__ float sD3[8][16][4];
  __shared__ int   sSrc[8][16];
  __shared__ float sW2[64], sW2b[8], sSc8[8], sOf8[8];
  __shared__ float shs[64], shq[64];
  if (t < 64) { sW2[t] = w2w[t]; shs[t] = 0.f; shq[t] = 0.f; }
  if (t < 8) { sW2b[t] = w2b[t]; sSc8[t] = sc8[t]; sOf8[t] = of8[t]; }

  const int cA = lane, cB = lane + 32;
  const float pA0 = p2w[cA * 3], pA1 = p2w[cA * 3 + 1], pA2 = p2w[cA * 3 + 2], pb0 = p2b[cA];
  const float pB0 = p2w[cB * 3], pB1 = p2w[cB * 3 + 1], pB2 = p2w[cB * 3 + 2], pb1 = p2b[cB];
  const float s30 = sc3[0], s31 = sc3[1], s32 = sc3[2];
  const float o30 = of3[0], o31 = of3[1], o32 = of3[2];
  __syncthreads();

  float st0 = 0, st1 = 0, sq0 = 0, sq1 = 0;
  const int stride = gridDim.x * 8;       // nodes per sweep (grid=1024 -> 8192)
  const int iters = NPTS / stride;        // uniform trip count for all waves
  for (int it = 0; it < iters; ++it) {
    const int node = it * stride + blockIdx.x * 8 + wv;
    const int e = lane & 15;              // lanes 16-31 mirror 0-15
    const int egi = node * 16 + e;

    // --- phase A: attention logits + softmax over the 16 edges -------------
    float aw[8];
    #pragma unroll
    for (int s = 0; s < 8; ++s) {
      float v = a2[(size_t)egi * 8 + s] * sSc8[s] + sOf8[s];
      aw[s] = v > 0.f ? v : 0.f;
    }
    #pragma unroll
    for (int s2 = 0; s2 < 8; ++s2) {
      float acc = sW2b[s2];
      #pragma unroll
      for (int s = 0; s < 8; ++s) acc += aw[s] * sW2[s2 * 8 + s];
      float mx = acc;
      #pragma unroll
      for (int d = 1; d < 16; d <<= 1) mx = fmaxf(mx, __shfl_xor(mx, d));
      float ex = __expf(acc - mx);
      float sm = ex;
      #pragma unroll
      for (int d = 1; d < 16; d <<= 1) sm += __shfl_xor(sm, d);
      if (lane < 16) sAttn[wv][e][s2] = ex / sm;
    }
    if (lane < 16) {
      sSrc[wv][e] = ei[egi];
      sD3[wv][e][0] = fmaxf(t3[(size_t)egi * 3 + 0] * s30 + o30, 0.f);
      sD3[wv][e][1] = fmaxf(t3[(size_t)egi * 3 + 1] * s31 + o31, 0.f);
      sD3[wv][e][2] = fmaxf(t3[(size_t)egi * 3 + 2] * s32 + o32, 0.f);
    }
    __syncthreads();

    // --- phase B: weighted aggregation, lane = channel ---------------------
    float acc0 = 0.f, acc1 = 0.f;
    #pragma unroll
    for (int ee = 0; ee < 16; ++ee) {
      int src = sSrc[wv][ee];
      float d0 = sD3[wv][ee][0], d1 = sD3[wv][ee][1], d2 = sD3[wv][ee][2];
      float at = sAttn[wv][ee][lane & 7];   // (lane+32)&7 == lane&7
      acc0 += at * (vb[(size_t)src * 64 + cA] + (pA0 * d0 + pA1 * d1 + pA2 * d2 + pb0));
      acc1 += at * (vb[(size_t)src * 64 + cB] + (pB0 * d0 + pB1 * d1 + pB2 * d2 + pb1));
    }
    outb[(size_t)node * 64 + cA] = acc0;
    outb[(size_t)node * 64 + cB] = acc1;
    st0 += acc0; sq0 += acc0 * acc0;
    st1 += acc1; sq1 += acc1 * acc1;
    __syncthreads();
  }
  atomicAdd(&shs[cA], st0); atomicAdd(&shs[cB], st1);
  atomicAdd(&shq[cA], sq0); atomicAdd(&shq[cB], sq1);
  __syncthreads();
  if (t < 64) { atomicAdd(&ssum[t], shs[t]); atomicAdd(&ssq[t], shq[t]); }
}

// final: out = relu(bn3(y3) + x_skip)
__global__ __launch_bounds__(256)
void final_kernel(const float* __restrict__ y3, const float* __restrict__ sc,
                  const float* __restrict__ of, const float* __restrict__ x,
                  float* __restrict__ out) {
  int i = blockIdx.x * 256 + threadIdx.x;
  int c = i & 63;
  float v = y3[i] * sc[c] + of[c] + x[i];
  out[i] = v > 0.f ? v : 0.f;
}

// ===========================================================================
extern "C" void kernel_launch(void* const* d_in, const int* in_sizes, int n_in,
                              void* d_out, int out_size, void* d_ws, size_t ws_size,
                              hipStream_t stream) {
  (void)in_sizes; (void)n_in; (void)out_size; (void)ws_size;
  const float* pos  = (const float*)d_in[0];
  const float* x    = (const float*)d_in[1];
  const float* W1   = (const float*)d_in[2];
  const float* Wk   = (const float*)d_in[3];
  const float* bk   = (const float*)d_in[4];
  const float* Wq   = (const float*)d_in[5];
  const float* bq   = (const float*)d_in[6];
  const float* Wv   = (const float*)d_in[7];
  const float* bv   = (const float*)d_in[8];
  const float* p1w  = (const float*)d_in[9];
  const float* p1b  = (const float*)d_in[10];
  const float* pbg  = (const float*)d_in[11];
  const float* pbb  = (const float*)d_in[12];
  const float* p2w  = (const float*)d_in[13];
  const float* p2b  = (const float*)d_in[14];
  const float* wb1g = (const float*)d_in[15];
  const float* wb1b = (const float*)d_in[16];
  const float* w1w  = (const float*)d_in[17];
  const float* w1b  = (const float*)d_in[18];
  const float* wb2g = (const float*)d_in[19];
  const float* wb2b = (const float*)d_in[20];
  const float* w2w  = (const float*)d_in[21];
  const float* w2b  = (const float*)d_in[22];
  const float* W3   = (const float*)d_in[23];
  const float* bn1g = (const float*)d_in[24];
  const float* bn1b = (const float*)d_in[25];
  const float* bn2g = (const float*)d_in[26];
  const float* bn2b = (const float*)d_in[27];
  const float* bn3g = (const float*)d_in[28];
  const float* bn3b = (const float*)d_in[29];
  const int*   ei   = (const int*)d_in[30];  // row 0 = src

  // ---- workspace carve-up (floats) ----
  float* ws = (float*)d_ws;
  const size_t NC = (size_t)NPTS * 64;
  float* Y1 = ws;                 // [N,64]  pre-BN linear1; reused as OUT
  float* H  = Y1 + NC;            // [N,64]  h; reused as h2
  float* KB = H + NC;             // [N,64]  k; reused as y3
  float* QB = KB + NC;            // [N,64]  q
  float* VB = QB + NC;            // [N,64]  v
  float* T3 = VB + NC;            // [E,3]
  float* A2 = T3 + (size_t)EDG * 3;  // [E,8]
  float* SB = A2 + (size_t)EDG * 8;  // stats block (2048 floats)
  float* OUTB = Y1;
  float* H2 = H;
  float* Y3 = KB;

  // zero BN-stat accumulators every call (deterministic under graph replay)
  hipMemsetAsync(SB, 0, 2048 * sizeof(float), stream);

  const float invN = 1.0f / (float)NPTS;
  const float invE = 1.0f / (float)EDG;

  // 1) y1 = x @ W1^T  (+ bn1 stats)
  gemm64_nt_kernel<<<NPTS / 64, 128, 0, stream>>>(x, W1, nullptr, Y1, SB + 0, SB + 64);
  finalize_kernel<<<1, 64, 0, stream>>>(SB + 0, SB + 64, bn1g, bn1b, invN, 64, SB + 128, SB + 192);
  bnrelu_kernel<<<(NPTS * 64) / 256, 256, 0, stream>>>(Y1, SB + 128, SB + 192, H);

  // 2) k, q, v
  gemm64_nt_kernel<<<NPTS / 64, 128, 0, stream>>>(H, Wk, bk, KB, nullptr, nullptr);
  gemm64_nt_kernel<<<NPTS / 64, 128, 0, stream>>>(H, Wq, bq, QB, nullptr, nullptr);
  gemm64_nt_kernel<<<NPTS / 64, 128, 0, stream>>>(H, Wv, bv, VB, nullptr, nullptr);

  // 3) positional branch t3 + stats
  edge_t3_kernel<<<EDG / 256, 256, 0, stream>>>(pos, ei, p1w, p1b, T3, SB + 256, SB + 264);
  finalize_kernel<<<1, 64, 0, stream>>>(SB + 256, SB + 264, pbg, pbb, invE, 3, SB + 272, SB + 280);

  // 4) stats of a = k[src]-q[dst]+delta  (bn_w1)
  edge_astats_kernel<<<2048, 256, 0, stream>>>(KB, QB, T3, SB + 272, SB + 280,
                                               p2w, p2b, ei, SB + 320, SB + 384);
  finalize_kernel<<<1, 64, 0, stream>>>(SB + 320, SB + 384, wb1g, wb1b, invE, 64, SB + 448, SB + 512);

  // 5) a2 = relu(bn_w1(a)) @ w1^T + w1_b  (+ bn_w2 stats)
  edge_a2_kernel<<<2048, 256, 0, stream>>>(KB, QB, T3, SB + 272, SB + 280,
                                           SB + 448, SB + 512, p2w, p2b, w1w, w1b,
                                           ei, A2, SB + 576, SB + 584);
  finalize_kernel<<<1, 64, 0, stream>>>(SB + 576, SB + 584, wb2g, wb2b, invE, 8, SB + 592, SB + 600);

  // 6) softmax + weighted aggregation (+ bn2 stats)
  node_aggr_kernel<<<1024, 256, 0, stream>>>(A2, SB + 592, SB + 600, w2w, w2b,
                                             T3, SB + 272, SB + 280, p2w, p2b,
                                             VB, ei, OUTB, SB + 640, SB + 704);
  finalize_kernel<<<1, 64, 0, stream>>>(SB + 640, SB + 704, bn2g, bn2b, invN, 64, SB + 768, SB + 832);
  bnrelu_kernel<<<(NPTS * 64) / 256, 256, 0, stream>>>(OUTB, SB + 768, SB + 832, H2);

  // 7) y3 = h2 @ W3^T (+ bn3 stats), residual + relu
  gemm64_nt_kernel<<<NPTS / 64, 128, 0, stream>>>(H2, W3, nullptr, Y3, SB + 896, SB + 960);
  finalize_kernel<<<1, 64, 0, stream>>>(SB + 896, SB + 960, bn3g, bn3b, invN, 64, SB + 1024, SB + 1088);
  final_kernel<<<(NPTS * 64) / 256, 256, 0, stream>>>(Y3, SB + 1024, SB + 1088, x, (float*)d_out);
}